// BiMambaRefinerBlock_56925496541907
// MI455X (gfx1250) — compile-verified
//
#include <hip/hip_runtime.h>
#include <hip/hip_bf16.h>
#include <math.h>
#include <stdint.h>

#define D_MODEL 768
#define D_INNER 1536
#define D_STATE 16
#define D_CONV  4
#define DT_RANK 48
#define BB 2
#define TT 4
#define NNT 196
#define LL (TT*NNT)          // 784
#define BL (BB*LL)           // 1568
#define XZ_N (2*D_INNER)     // 3072
#define XDBL_N (DT_RANK + 2*D_STATE) // 80

typedef float v2f __attribute__((ext_vector_type(2)));
typedef float v8f __attribute__((ext_vector_type(8)));

__device__ __forceinline__ int mapRow(int m) {
  int b = m / LL, l = m % LL;
  int t = l / NNT, n = l % NNT;
  return b * LL + (TT - 1 - t) * NNT + n;
}
__device__ __forceinline__ float sigmoidf_(float v) { return 1.f / (1.f + __expf(-v)); }
__device__ __forceinline__ float siluf_(float v) { return v * sigmoidf_(v); }

// async global -> LDS copy of 4 bytes (CDNA5, ASYNCcnt-tracked), GVS mode:
// LDS byte address ldsOff; global address base + gByteOff
__device__ __forceinline__ void async_g2l_b32(uint32_t ldsOff, uint32_t gByteOff,
                                              const float* base) {
  asm volatile("global_load_async_to_lds_b32 %0, %1, %2"
               :: "v"(ldsOff), "v"(gByteOff), "s"(base) : "memory");
}
__device__ __forceinline__ void wait_async0() {
  asm volatile("s_wait_asynccnt 0x0" ::: "memory");
}

// ---------------------------------------------------------------------------
// RMSNorm with optional time-flip gather (writes in the flipped domain).
// ---------------------------------------------------------------------------
__global__ void rmsnorm_map_kernel(const float* __restrict__ x,
                                   const float* __restrict__ w,
                                   float* __restrict__ xn, int flip) {
  int m = blockIdx.x;                       // destination row (flipped domain)
  int src = flip ? mapRow(m) : m;
  const float* xr = x + (size_t)src * D_MODEL;
  float v0 = xr[threadIdx.x];
  float v1 = xr[threadIdx.x + 256];
  float v2 = xr[threadIdx.x + 512];
  float s = v0 * v0 + v1 * v1 + v2 * v2;
  __shared__ float red[256];
  red[threadIdx.x] = s;
  __syncthreads();
  for (int o = 128; o > 0; o >>= 1) {
    if (threadIdx.x < o) red[threadIdx.x] += red[threadIdx.x + o];
    __syncthreads();
  }
  float scale = rsqrtf(red[0] * (1.f / D_MODEL) + 1e-5f);
  float* xo = xn + (size_t)m * D_MODEL;
  xo[threadIdx.x]        = v0 * scale * w[threadIdx.x];
  xo[threadIdx.x + 256]  = v1 * scale * w[threadIdx.x + 256];
  xo[threadIdx.x + 512]  = v2 * scale * w[threadIdx.x + 512];
}

// ---------------------------------------------------------------------------
// WMMA f32 GEMM: C[m,n] = sum_k A[m,k]*W[n,k]  (W row-major N x K)
// 256 threads = 8 waves (4x2 grid of 16x16 tiles) => 64x32 macro-tile.
// Tile staging: async global->LDS (full tiles) / guarded zero-fill (edges).
// MODE: 0 plain, 1 softplus(+bias), 2 residual+flip-map store,
//       3 sigmoid gate combine(+bias), 4 bias add
// ---------------------------------------------------------------------------
template <int MODE>
__global__ void gemm_wmma_kernel(const float* __restrict__ A, int lda,
                                 const float* __restrict__ W, int ldw,
                                 float* __restrict__ Cout, int ldc, int ccol,
                                 int M, int N, int K,
                                 const float* __restrict__ bias,
                                 const float* __restrict__ xref,
                                 int flip) {
  __shared__ float As[32][64 + 1];   // K-major: As[k][m], pitch 65 (banks coprime)
  __shared__ float Bs[32][32 + 1];   // K-major: Bs[k][n], pitch 33
  int m_tile = blockIdx.y * 64;
  int n_tile = blockIdx.x * 32;
  int tid = threadIdx.x;
  int wave = tid >> 5, lane = tid & 31;
  int wr = wave >> 1, wc = wave & 1;
  int wm = wr * 16, wn = wc * 16;
  int half = lane >> 4, idx = lane & 15;
  uint32_t as0 = (uint32_t)(uintptr_t)&As[0][0];
  uint32_t bs0 = (uint32_t)(uintptr_t)&Bs[0][0];
  bool mfull = (m_tile + 64 <= M);
  bool nfull = (n_tile + 32 <= N);
  v8f acc = {0.f, 0.f, 0.f, 0.f, 0.f, 0.f, 0.f, 0.f};

  for (int k0 = 0; k0 < K; k0 += 32) {
    bool kfull = (k0 + 32 <= K);
    if (mfull && kfull) {
      // async DMA: global -> LDS, no VGPR round-trip (ASYNCcnt)
#pragma unroll
      for (int r = 0; r < 8; ++r) {
        int i = tid + r * 256;
        int mm = i >> 5, kk = i & 31;
        uint32_t ldsa = as0 + (uint32_t)(kk * 65 + mm) * 4u;
        uint32_t goff = (uint32_t)((m_tile + mm) * lda + (k0 + kk)) * 4u;
        async_g2l_b32(ldsa, goff, A);
      }
    } else {
      for (int i = tid; i < 64 * 32; i += 256) {
        int mm = i >> 5, kk = i & 31;
        int gm = m_tile + mm, gk = k0 + kk;
        As[kk][mm] = (gm < M && gk < K) ? A[(size_t)gm * lda + gk] : 0.f;
      }
    }
    if (nfull && kfull) {
#pragma unroll
      for (int r = 0; r < 4; ++r) {
        int i = tid + r * 256;
        int nn = i >> 5, kk = i & 31;
        uint32_t ldsa = bs0 + (uint32_t)(kk * 33 + nn) * 4u;
        uint32_t goff = (uint32_t)((n_tile + nn) * ldw + (k0 + kk)) * 4u;
        async_g2l_b32(ldsa, goff, W);
      }
    } else {
      for (int i = tid; i < 32 * 32; i += 256) {
        int nn = i >> 5, kk = i & 31;
        int gn = n_tile + nn, gk = k0 + kk;
        Bs[kk][nn] = (gn < N && gk < K) ? W[(size_t)gn * ldw + gk] : 0.f;
      }
    }
    wait_async0();        // retire this wave's async copies before publishing
    __syncthreads();
#pragma unroll
    for (int k4 = 0; k4 < 32; k4 += 4) {
      // ISA 7.12.2 32-bit A/B layout: lanes 0-15 -> K+0/K+1, lanes 16-31 -> K+2/K+3
      v2f a, b;
      a.x = As[k4 + 2 * half + 0][wm + idx];
      a.y = As[k4 + 2 * half + 1][wm + idx];
      b.x = Bs[k4 + 2 * half + 0][wn + idx];
      b.y = Bs[k4 + 2 * half + 1][wn + idx];
      acc = __builtin_amdgcn_wmma_f32_16x16x4_f32(false, a, false, b,
                                                  (short)0, acc, false, false);
    }
    __syncthreads();
  }

  // C/D layout: VGPR r, lanes 0-15 -> M=r, lanes 16-31 -> M=r+8, N=idx
  int gn = n_tile + wn + idx;
#pragma unroll
  for (int r = 0; r < 8; ++r) {
    int gm = m_tile + wm + half * 8 + r;
    if (gm < M && gn < N) {
      float v = acc[r];
      if (MODE == 0) {
        Cout[(size_t)gm * ldc + ccol + gn] = v;
      } else if (MODE == 1) {               // dt = softplus(v + b)
        v += bias[gn];
        v = (v > 20.f) ? v : log1pf(__expf(v));
        Cout[(size_t)gm * ldc + ccol + gn] = v;
      } else if (MODE == 2) {               // residual + un-flip store
        int dm = flip ? mapRow(gm) : gm;
        Cout[(size_t)dm * ldc + ccol + gn] = v + xref[(size_t)dm * D_MODEL + gn];
      } else if (MODE == 3) {               // gate combine
        float g = sigmoidf_(v + bias[gn]);
        float of = xref[(size_t)gm * (2 * D_MODEL) + gn];
        float ob = xref[(size_t)gm * (2 * D_MODEL) + D_MODEL + gn];
        Cout[(size_t)gm * ldc + gn] = g * of + (1.f - g) * ob;
      } else {                              // MODE 4: bias
        Cout[(size_t)gm * ldc + gn] = v + bias[gn];
      }
    }
  }
}

// ---------------------------------------------------------------------------
// Causal depthwise conv (k=4) + bias + SiLU, reading xb half of xz.
// ---------------------------------------------------------------------------
__global__ void conv_silu_kernel(const float* __restrict__ xz,
                                 const float* __restrict__ cw,
                                 const float* __restrict__ cb,
                                 float* __restrict__ xbc) {
  int i = blockIdx.x * 256 + threadIdx.x;
  if (i >= BL * D_INNER) return;
  int d = i % D_INNER;
  int m = i / D_INNER;
  int l = m % LL;
  float acc = cb[d];
  float w0 = cw[d * 4 + 0], w1 = cw[d * 4 + 1], w2 = cw[d * 4 + 2], w3 = cw[d * 4 + 3];
  if (l >= 3) acc += xz[(size_t)(m - 3) * XZ_N + d] * w0;
  if (l >= 2) acc += xz[(size_t)(m - 2) * XZ_N + d] * w1;
  if (l >= 1) acc += xz[(size_t)(m - 1) * XZ_N + d] * w2;
  acc += xz[(size_t)m * XZ_N + d] * w3;
  xbc[(size_t)m * D_INNER + d] = siluf_(acc);
}

// ---------------------------------------------------------------------------
// Selective scan: one (b,d) chain per lane, h[16] in registers,
// B/C vectors (32 floats/step) staged through LDS. Fuses *silu(z) and +u*D.
// Prefetches the next step's dt/u rows (serial latency bottleneck).
// ---------------------------------------------------------------------------
__global__ void scan_kernel(const float* __restrict__ u,     // xbc
                            const float* __restrict__ dtb,
                            const float* __restrict__ xdbl,
                            const float* __restrict__ xz,    // for z half
                            const float* __restrict__ A_log,
                            const float* __restrict__ Dp,
                            float* __restrict__ y) {
  int b = blockIdx.x / (D_INNER / 256);
  int d = (blockIdx.x % (D_INNER / 256)) * 256 + threadIdx.x;
  float A[D_STATE], h[D_STATE];
#pragma unroll
  for (int n = 0; n < D_STATE; ++n) {
    A[n] = -__expf(A_log[(size_t)d * D_STATE + n]);
    h[n] = 0.f;
  }
  float Dd = Dp[d];
  __shared__ float sBC[32];
  for (int l = 0; l < LL; ++l) {
    int m = b * LL + l;
    if (threadIdx.x < 32)
      sBC[threadIdx.x] = xdbl[(size_t)m * XDBL_N + DT_RANK + threadIdx.x];
    __syncthreads();
    if (l + 1 < LL) {   // hide next-step load latency (gfx1250 global_prefetch)
      __builtin_prefetch(&dtb[(size_t)(m + 1) * D_INNER + d], 0, 0);
      __builtin_prefetch(&u[(size_t)(m + 1) * D_INNER + d], 0, 0);
    }
    float dt = dtb[(size_t)m * D_INNER + d];
    float uu = u[(size_t)m * D_INNER + d];
    float du = dt * uu;
    float yy = 0.f;
#pragma unroll
    for (int n = 0; n < D_STATE; ++n) {
      h[n] = __expf(dt * A[n]) * h[n] + du * sBC[n];
      yy += h[n] * sBC[D_STATE + n];
    }
    float z = xz[(size_t)m * XZ_N + D_INNER + d];
    y[(size_t)m * D_INNER + d] = (yy + uu * Dd) * siluf_(z);
    __syncthreads();
  }
}

// ---------------------------------------------------------------------------
extern "C" void kernel_launch(void* const* d_in, const int* in_sizes, int n_in,
                              void* d_out, int out_size, void* d_ws, size_t ws_size,
                              hipStream_t stream) {
  const float* x      = (const float*)d_in[0];
  const float* norm_w = (const float*)d_in[1];
  const float* in_w   = (const float*)d_in[2];
  const float* conv_w = (const float*)d_in[3];
  const float* conv_b = (const float*)d_in[4];
  const float* xp_w   = (const float*)d_in[5];
  const float* dt_w   = (const float*)d_in[6];
  const float* dt_b   = (const float*)d_in[7];
  const float* A_log  = (const float*)d_in[8];
  const float* Dp     = (const float*)d_in[9];
  const float* mo_w   = (const float*)d_in[10];
  const float* gate_w = (const float*)d_in[11];
  const float* gate_b = (const float*)d_in[12];
  const float* proj_w = (const float*)d_in[13];
  const float* proj_b = (const float*)d_in[14];
  float* out = (float*)d_out;

  float* w    = (float*)d_ws;
  float* xn   = w;                           // BL*768
  float* xz   = xn   + (size_t)BL * D_MODEL; // BL*3072
  float* xbc  = xz   + (size_t)BL * XZ_N;    // BL*1536
  float* xdbl = xbc  + (size_t)BL * D_INNER; // BL*80
  float* dtv  = xdbl + (size_t)BL * XDBL_N;  // BL*1536
  float* yb   = dtv  + (size_t)BL * D_INNER; // BL*1536
  float* ocat = yb   + (size_t)BL * D_INNER; // BL*1536 (out_f | out_b)
  float* comb = ocat + (size_t)BL * D_INNER; // BL*768

  dim3 blk(256);
  int gy = (BL + 63) / 64;  // 25

  for (int dir = 0; dir < 2; ++dir) {
    rmsnorm_map_kernel<<<BL, 256, 0, stream>>>(x, norm_w + dir * D_MODEL, xn, dir);

    // in_proj: (BL x 3072) = xn (BL x 768) * in_w^T
    gemm_wmma_kernel<0><<<dim3(XZ_N / 32, gy), blk, 0, stream>>>(
        xn, D_MODEL, in_w + (size_t)dir * XZ_N * D_MODEL, D_MODEL,
        xz, XZ_N, 0, BL, XZ_N, D_MODEL, nullptr, nullptr, 0);

    conv_silu_kernel<<<(BL * D_INNER + 255) / 256, 256, 0, stream>>>(
        xz, conv_w + dir * D_INNER * D_CONV, conv_b + dir * D_INNER, xbc);

    // x_proj: (BL x 80) = xbc (BL x 1536) * xp_w^T
    gemm_wmma_kernel<0><<<dim3((XDBL_N + 31) / 32, gy), blk, 0, stream>>>(
        xbc, D_INNER, xp_w + (size_t)dir * XDBL_N * D_INNER, D_INNER,
        xdbl, XDBL_N, 0, BL, XDBL_N, D_INNER, nullptr, nullptr, 0);

    // dt_proj + softplus: (BL x 1536) = xdbl[:, :48] * dt_w^T + dt_b
    gemm_wmma_kernel<1><<<dim3(D_INNER / 32, gy), blk, 0, stream>>>(
        xdbl, XDBL_N, dt_w + (size_t)dir * D_INNER * DT_RANK, DT_RANK,
        dtv, D_INNER, 0, BL, D_INNER, DT_RANK, dt_b + dir * D_INNER, nullptr, 0);

    scan_kernel<<<BB * (D_INNER / 256), 256, 0, stream>>>(
        xbc, dtv, xdbl, xz, A_log + (size_t)dir * D_INNER * D_STATE,
        Dp + dir * D_INNER, yb);

    // mix_out + residual, un-flip into concat buffer column block
    gemm_wmma_kernel<2><<<dim3(D_MODEL / 32, gy), blk, 0, stream>>>(
        yb, D_INNER, mo_w + (size_t)dir * D_MODEL * D_INNER, D_INNER,
        ocat, 2 * D_MODEL, dir * D_MODEL, BL, D_MODEL, D_INNER, nullptr, x, dir);
  }

  // gate = sigmoid(cat * gate_w^T + gate_b); comb = g*out_f + (1-g)*out_b
  gemm_wmma_kernel<3><<<dim3(D_MODEL / 32, gy), blk, 0, stream>>>(
      ocat, 2 * D_MODEL, gate_w, 2 * D_MODEL,
      comb, D_MODEL, 0, BL, D_MODEL, 2 * D_MODEL, gate_b, ocat, 0);

  // final projection + bias -> d_out
  gemm_wmma_kernel<4><<<dim3(D_MODEL / 32, gy), blk, 0, stream>>>(
      comb, D_MODEL, proj_w, D_MODEL,
      out, D_MODEL, 0, BL, D_MODEL, D_MODEL, proj_b, nullptr, 0);
}